// GraphSAGE_GAT_5738076307732
// MI455X (gfx1250) — compile-verified
//
#include <hip/hip_runtime.h>
#include <math.h>

typedef __attribute__((ext_vector_type(2))) float v2f;
typedef __attribute__((ext_vector_type(8))) float v8f;

constexpr long long NN = 50000;   // nodes
constexpr long long NE = 800000;  // edges
constexpr int IN = 16, EF = 8, HID = 128, HEADS = 4, ACT = 32, G = 64;
constexpr float NEG_SLOPE = 0.2f;

// ---------------- utility ----------------
__global__ void k_fill(float* __restrict__ p, long long n, float v) {
  long long i = (long long)blockIdx.x * blockDim.x + threadIdx.x;
  if (i < n) p[i] = v;
}

__device__ inline void atomicMaxF(float* addr, float val) {
  if (val >= 0.0f) atomicMax((int*)addr, __float_as_int(val));
  else             atomicMin((unsigned int*)addr, (unsigned int)__float_as_int(val));
}

// ---------------- degree ----------------
__global__ void k_degree(const int* __restrict__ ei, float* __restrict__ deg) {
  long long e = (long long)blockIdx.x * blockDim.x + threadIdx.x;
  if (e < NE) atomicAdd(&deg[ei[NE + e]], 1.0f);
}

// ---------------- NNConv fused edge kernel ----------------
// 16 threads per edge (one per output dim), 128 edges per block.
constexpr int EPB = 128;
__global__ __launch_bounds__(256)
void k_nnconv(const float* __restrict__ x, const int* __restrict__ ei,
              const float* __restrict__ ea,
              const float* __restrict__ w1, const float* __restrict__ b1,
              const float* __restrict__ w2, const float* __restrict__ b2,
              float* __restrict__ agg1) {
  __shared__ float s_w2[32 * 256];
  __shared__ float s_w1[8 * 32];
  __shared__ float s_b2[256];
  __shared__ float s_b1[32];
  __shared__ float s_h[16][33];
  __shared__ float s_xs[16][17];
  const int tid = threadIdx.x;
  for (int i = tid; i < 32 * 256; i += 256) s_w2[i] = w2[i];
  for (int i = tid; i < 8 * 32; i += 256)   s_w1[i] = w1[i];
  if (tid < 256) s_b2[tid] = b2[tid];
  if (tid < 32)  s_b1[tid] = b1[tid];
  __syncthreads();

  const int team = tid >> 4;     // 0..15 (edge within round)
  const int o    = tid & 15;     // output dim
  const long long base = (long long)blockIdx.x * EPB;
  for (int it = 0; it < EPB / 16; ++it) {
    const long long e = base + (long long)it * 16 + team;
    const bool valid = (e < NE);
    int dst = 0;
    if (valid) {
      const int src = ei[e];
      dst = ei[NE + e];
      float a0 = s_b1[o], a1 = s_b1[o + 16];
#pragma unroll
      for (int k = 0; k < EF; ++k) {
        const float v = ea[e * EF + k];
        a0 += v * s_w1[k * 32 + o];
        a1 += v * s_w1[k * 32 + o + 16];
      }
      s_h[team][o]      = fmaxf(a0, 0.0f);
      s_h[team][o + 16] = fmaxf(a1, 0.0f);
      s_xs[team][o] = x[(long long)src * IN + o];
    }
    __syncthreads();
    if (valid) {
      float m = 0.0f;
#pragma unroll
      for (int i = 0; i < IN; ++i) {
        float w = s_b2[i * 16 + o];
        const float* wcol = &s_w2[i * 16 + o];
#pragma unroll 8
        for (int k = 0; k < 32; ++k) w += s_h[team][k] * wcol[k * 256];
        m += s_xs[team][i] * w;
      }
      atomicAdd(&agg1[(long long)dst * IN + o], m);
    }
    __syncthreads();
  }
}

// ---------------- x1 = relu(agg1/deg + x@nn_root + bias) ----------------
__global__ void k_x1(const float* __restrict__ agg1, const float* __restrict__ deg,
                     const float* __restrict__ x, const float* __restrict__ nn_root,
                     const float* __restrict__ nn_bias, float* __restrict__ x1) {
  long long gid = (long long)blockIdx.x * blockDim.x + threadIdx.x;
  if (gid >= NN * IN) return;
  const long long n = gid >> 4;
  const int o = (int)(gid & 15);
  float acc = agg1[gid] / fmaxf(deg[n], 1.0f) + nn_bias[o];
#pragma unroll
  for (int i = 0; i < IN; ++i) acc += x[n * IN + i] * nn_root[i * IN + o];
  x1[gid] = fmaxf(acc, 0.0f);
}

// ---------------- SAGE gather ----------------
__global__ void k_sage_gather(const int* __restrict__ ei, const float* __restrict__ x1,
                              float* __restrict__ agg2) {
  long long gid = (long long)blockIdx.x * blockDim.x + threadIdx.x;
  if (gid >= NE * IN) return;
  const long long e = gid >> 4;
  const int i = (int)(gid & 15);
  atomicAdd(&agg2[(long long)ei[NE + e] * IN + i], x1[(long long)ei[e] * IN + i]);
}

// ---------------- SAGE GEMM via WMMA f32 16x16x4 ----------------
__global__ __launch_bounds__(128)
void k_sage(const float* __restrict__ agg2, const float* __restrict__ x1,
            const float* __restrict__ deg,
            const float* __restrict__ wl, const float* __restrict__ wr,
            const float* __restrict__ bl, float* __restrict__ x2) {
  const int lane = threadIdx.x;
  const int half = lane >> 4;
  const int lrow = lane & 15;
  const int row0 = blockIdx.x * 16;
  const int col0 = (blockIdx.y * 4 + threadIdx.y) * 16;
  const int row = row0 + lrow;
  const float invd = 1.0f / fmaxf(deg[row], 1.0f);
  v8f c = {};
#pragma unroll
  for (int k0 = 0; k0 < IN; k0 += 4) {
    const int ka = k0 + 2 * half;
    v2f a, b;
    a[0] = agg2[(long long)row * IN + ka]     * invd;
    a[1] = agg2[(long long)row * IN + ka + 1] * invd;
    b[0] = wl[ka * HID + col0 + lrow];
    b[1] = wl[(ka + 1) * HID + col0 + lrow];
    c = __builtin_amdgcn_wmma_f32_16x16x4_f32(false, a, false, b, (short)0, c, false, false);
    a[0] = x1[(long long)row * IN + ka];
    a[1] = x1[(long long)row * IN + ka + 1];
    b[0] = wr[ka * HID + col0 + lrow];
    b[1] = wr[(ka + 1) * HID + col0 + lrow];
    c = __builtin_amdgcn_wmma_f32_16x16x4_f32(false, a, false, b, (short)0, c, false, false);
  }
  const float bias = bl[col0 + lrow];
#pragma unroll
  for (int r = 0; r < 8; ++r) {
    const int m = row0 + r + 8 * half;
    x2[(long long)m * HID + col0 + lrow] = fmaxf(c[r] + bias, 0.0f);
  }
}

// ---------------- GAT projection GEMM via WMMA: xp = x2 @ gat_w ----------------
__global__ __launch_bounds__(128)
void k_gat_xp(const float* __restrict__ x2, const float* __restrict__ gw,
              float* __restrict__ xp) {
  const int lane = threadIdx.x;
  const int half = lane >> 4;
  const int lrow = lane & 15;
  const int row = blockIdx.x * 16 + lrow;
  const int col0 = (blockIdx.y * 4 + threadIdx.y) * 16;
  v8f c = {};
#pragma unroll 8
  for (int k0 = 0; k0 < HID; k0 += 4) {
    const int ka = k0 + 2 * half;
    v2f a, b;
    a[0] = x2[(long long)row * HID + ka];
    a[1] = x2[(long long)row * HID + ka + 1];
    b[0] = gw[ka * 512 + col0 + lrow];
    b[1] = gw[(ka + 1) * 512 + col0 + lrow];
    c = __builtin_amdgcn_wmma_f32_16x16x4_f32(false, a, false, b, (short)0, c, false, false);
  }
#pragma unroll
  for (int r = 0; r < 8; ++r) {
    const int m = blockIdx.x * 16 + r + 8 * half;
    xp[(long long)m * 512 + col0 + lrow] = c[r];
  }
}

// ---------------- attention coefficients ----------------
__global__ void k_attn(const float* __restrict__ xp, const float* __restrict__ att_s,
                       const float* __restrict__ att_d,
                       float* __restrict__ a_src, float* __restrict__ a_dst) {
  long long gid = (long long)blockIdx.x * blockDim.x + threadIdx.x;
  if (gid >= NN * HEADS) return;
  const long long n = gid >> 2;
  const int h = (int)(gid & 3);
  const float* xr = xp + n * 512 + (long long)h * HID;
  float s1 = 0.0f, s2 = 0.0f;
#pragma unroll 8
  for (int d = 0; d < HID; ++d) {
    const float v = xr[d];
    s1 += v * att_s[h * HID + d];
    s2 += v * att_d[h * HID + d];
  }
  a_src[gid] = s1;
  a_dst[gid] = s2;
}

__device__ inline void edge_pair(long long idx, const int* __restrict__ ei,
                                 int& si, int& di) {
  if (idx < NE) { si = ei[idx]; di = ei[NE + idx]; }
  else          { si = di = (int)(idx - NE); }
}

// ---------------- edge softmax: pass 1 (max) ----------------
__global__ void k_edge_max(const int* __restrict__ ei, const float* __restrict__ a_src,
                           const float* __restrict__ a_dst, float* __restrict__ emax) {
  long long idx = (long long)blockIdx.x * blockDim.x + threadIdx.x;
  if (idx >= NE + NN) return;
  int si, di; edge_pair(idx, ei, si, di);
#pragma unroll
  for (int h = 0; h < HEADS; ++h) {
    float v = a_src[(long long)si * 4 + h] + a_dst[(long long)di * 4 + h];
    v = (v > 0.0f) ? v : NEG_SLOPE * v;
    atomicMaxF(&emax[(long long)di * 4 + h], v);
  }
}

// ---------------- edge softmax: pass 2 (exp-sum) ----------------
__global__ void k_edge_expsum(const int* __restrict__ ei, const float* __restrict__ a_src,
                              const float* __restrict__ a_dst, const float* __restrict__ emax,
                              float* __restrict__ denom) {
  long long idx = (long long)blockIdx.x * blockDim.x + threadIdx.x;
  if (idx >= NE + NN) return;
  int si, di; edge_pair(idx, ei, si, di);
#pragma unroll
  for (int h = 0; h < HEADS; ++h) {
    float v = a_src[(long long)si * 4 + h] + a_dst[(long long)di * 4 + h];
    v = (v > 0.0f) ? v : NEG_SLOPE * v;
    atomicAdd(&denom[(long long)di * 4 + h], __expf(v - emax[(long long)di * 4 + h]));
  }
}

// ---------------- edge softmax: pass 3 (weighted scatter) ----------------
__global__ void k_edge_accum(const int* __restrict__ ei, const float* __restrict__ a_src,
                             const float* __restrict__ a_dst, const float* __restrict__ emax,
                             const float* __restrict__ denom, const float* __restrict__ xp,
                             float* __restrict__ outacc) {
  long long gid = (long long)blockIdx.x * blockDim.x + threadIdx.x;
  if (gid >= (NE + NN) * (long long)HID) return;
  const long long idx = gid >> 7;
  const int d = (int)(gid & 127);
  int si, di; edge_pair(idx, ei, si, di);
#pragma unroll
  for (int h = 0; h < HEADS; ++h) {
    float v = a_src[(long long)si * 4 + h] + a_dst[(long long)di * 4 + h];
    v = (v > 0.0f) ? v : NEG_SLOPE * v;
    const float ex = __expf(v - emax[(long long)di * 4 + h]);
    const float alpha = ex / denom[(long long)di * 4 + h];
    atomicAdd(&outacc[(long long)di * 512 + h * HID + d],
              alpha * xp[(long long)si * 512 + h * HID + d]);
  }
}

// ---------------- x3 = relu(mean_heads(out) + gat_bias) ----------------
__global__ void k_x3(const float* __restrict__ outacc, const float* __restrict__ gat_bias,
                     float* __restrict__ x3) {
  long long gid = (long long)blockIdx.x * blockDim.x + threadIdx.x;
  if (gid >= NN * HID) return;
  const long long n = gid >> 7;
  const int d = (int)(gid & 127);
  const float* o = outacc + n * 512 + d;
  const float s = 0.25f * (o[0] + o[128] + o[256] + o[384]) + gat_bias[d];
  x3[gid] = fmaxf(s, 0.0f);
}

// ---------------- pooling ----------------
__global__ void k_count(const int* __restrict__ batch, float* __restrict__ gcnt) {
  long long n = (long long)blockIdx.x * blockDim.x + threadIdx.x;
  if (n < NN) atomicAdd(&gcnt[batch[n]], 1.0f);
}

__global__ void k_pool(const int* __restrict__ batch, const float* __restrict__ x3,
                       float* __restrict__ gsum) {
  long long gid = (long long)blockIdx.x * blockDim.x + threadIdx.x;
  if (gid >= NN * HID) return;
  const long long n = gid >> 7;
  const int d = (int)(gid & 127);
  atomicAdd(&gsum[(long long)batch[n] * HID + d], x3[gid]);
}

// ---------------- head ----------------
__global__ void k_head(const float* __restrict__ gsum, const float* __restrict__ gcnt,
                       const float* __restrict__ hw, const float* __restrict__ hb,
                       float* __restrict__ out) {
  int gid = blockIdx.x * blockDim.x + threadIdx.x;
  if (gid >= G * ACT) return;
  const int g = gid >> 5;
  const int a = gid & 31;
  const float inv = 1.0f / fmaxf(gcnt[g], 1.0f);
  float acc = hb[a];
#pragma unroll 8
  for (int d = 0; d < HID; ++d) acc += gsum[g * HID + d] * inv * hw[d * ACT + a];
  out[gid] = acc;
}

// ---------------- launch ----------------
static inline unsigned nb(long long n, int b) { return (unsigned)((n + b - 1) / b); }

extern "C" void kernel_launch(void* const* d_in, const int* in_sizes, int n_in,
                              void* d_out, int out_size, void* d_ws, size_t ws_size,
                              hipStream_t stream) {
  const float* x        = (const float*)d_in[0];
  const int*   ei       = (const int*)  d_in[1];
  const float* ea       = (const float*)d_in[2];
  const int*   batch    = (const int*)  d_in[3];
  const float* enc_w1   = (const float*)d_in[4];
  const float* enc_b1   = (const float*)d_in[5];
  const float* enc_w2   = (const float*)d_in[6];
  const float* enc_b2   = (const float*)d_in[7];
  const float* nn_root  = (const float*)d_in[8];
  const float* nn_bias  = (const float*)d_in[9];
  const float* sage_wl  = (const float*)d_in[10];
  const float* sage_bl  = (const float*)d_in[11];
  const float* sage_wr  = (const float*)d_in[12];
  const float* gat_w    = (const float*)d_in[13];
  const float* att_src  = (const float*)d_in[14];
  const float* att_dst  = (const float*)d_in[15];
  const float* gat_bias = (const float*)d_in[16];
  const float* head_w   = (const float*)d_in[17];
  const float* head_b   = (const float*)d_in[18];
  float* out = (float*)d_out;

  float* ws = (float*)d_ws;
  float* deg    = ws;                 // N
  float* a_src  = ws + NN;            // 4N
  float* a_dst  = ws + 5 * NN;        // 4N
  float* emax   = ws + 9 * NN;        // 4N
  float* denom  = ws + 13 * NN;       // 4N
  float* xp     = ws + 17 * NN;       // 512N ; x3 aliases (xp dead before x3 written)
  float* x3     = xp;                 // 128N
  float* scr    = ws + 529 * NN;      // 512N shared region:
  float* agg1   = scr;                //   16N   (dead after x1)
  float* x1     = scr + 16 * NN;      //   16N   (dead after x2)
  float* agg2   = scr + 32 * NN;      //   16N   (dead after x2)
  float* x2     = scr + 48 * NN;      //   128N  (dead after xp)
  float* outacc = scr;                //   512N  (reuses whole region afterwards)
  float* gsum   = ws + 1041 * NN;     // 128*G
  float* gcnt   = gsum + (long long)HID * G;  // G

  // init
  k_fill<<<nb(NN, 256), 256, 0, stream>>>(deg, NN, 0.0f);
  k_fill<<<nb(16 * NN, 256), 256, 0, stream>>>(agg1, 16 * NN, 0.0f);
  k_fill<<<nb(16 * NN, 256), 256, 0, stream>>>(agg2, 16 * NN, 0.0f);

  // degrees
  k_degree<<<nb(NE, 256), 256, 0, stream>>>(ei, deg);

  // NNConv
  k_nnconv<<<nb(NE, EPB), 256, 0, stream>>>(x, ei, ea, enc_w1, enc_b1, enc_w2, enc_b2, agg1);
  k_x1<<<nb(NN * IN, 256), 256, 0, stream>>>(agg1, deg, x, nn_root, nn_bias, x1);

  // SAGE
  k_sage_gather<<<nb(NE * IN, 256), 256, 0, stream>>>(ei, x1, agg2);
  k_sage<<<dim3((unsigned)(NN / 16), 2), dim3(32, 4), 0, stream>>>(agg2, x1, deg,
                                                                   sage_wl, sage_wr, sage_bl, x2);

  // GAT projection (WMMA GEMM)
  k_gat_xp<<<dim3((unsigned)(NN / 16), 8), dim3(32, 4), 0, stream>>>(x2, gat_w, xp);
  k_attn<<<nb(NN * HEADS, 256), 256, 0, stream>>>(xp, att_src, att_dst, a_src, a_dst);

  // edge softmax (x2 region now dead -> outacc reuses it)
  k_fill<<<nb(4 * NN, 256), 256, 0, stream>>>(emax, 4 * NN, -INFINITY);
  k_fill<<<nb(4 * NN, 256), 256, 0, stream>>>(denom, 4 * NN, 0.0f);
  k_fill<<<nb(512 * NN, 256), 256, 0, stream>>>(outacc, 512 * NN, 0.0f);
  k_edge_max<<<nb(NE + NN, 256), 256, 0, stream>>>(ei, a_src, a_dst, emax);
  k_edge_expsum<<<nb(NE + NN, 256), 256, 0, stream>>>(ei, a_src, a_dst, emax, denom);
  k_edge_accum<<<nb((NE + NN) * (long long)HID, 256), 256, 0, stream>>>(
      ei, a_src, a_dst, emax, denom, xp, outacc);

  // head-mean + bias + relu (xp dead -> x3 aliases xp)
  k_x3<<<nb(NN * HID, 256), 256, 0, stream>>>(outacc, gat_bias, x3);

  // pool + action head
  k_fill<<<nb((long long)HID * G + G, 256), 256, 0, stream>>>(gsum, (long long)HID * G + G, 0.0f);
  k_count<<<nb(NN, 256), 256, 0, stream>>>(batch, gcnt);
  k_pool<<<nb(NN * HID, 256), 256, 0, stream>>>(batch, x3, gsum);
  k_head<<<nb(G * ACT, 256), 256, 0, stream>>>(gsum, gcnt, head_w, head_b, out);
}